// GAE_46943992545743
// MI455X (gfx1250) — compile-verified
//
#include <hip/hip_runtime.h>

// ---------------- problem constants ----------------
#define U_N   8192
#define I_N   2048
#define R_N   5
#define DIN_N 256
#define MSK_N 512
#define AGG_NW 8   // waves per block in the aggregation kernels (split-K)

typedef float v2f __attribute__((ext_vector_type(2)));
typedef float v8f __attribute__((ext_vector_type(8)));

// ---------------- WMMA helpers (V_WMMA_F32_16X16X4_F32) ----------------
__device__ __forceinline__ v8f wmma_f32(v2f a, v2f b, v8f c) {
  // 8 args: (neg_a, A, neg_b, B, c_mod, C, reuse_a, reuse_b)
  return __builtin_amdgcn_wmma_f32_16x16x4_f32(false, a, false, b, (short)0, c,
                                               false, false);
}

// A fragment: 16x4 tile of row-major X (leading dim ld), tile origin (row0,k0).
// lane: m = L%16 row, h = L/16 selects K pair {2h, 2h+1}.
__device__ __forceinline__ v2f ldA(const float* X, int ld, int row0, int k0,
                                   int m, int h) {
  const float* p = X + (size_t)(row0 + m) * ld + (k0 + 2 * h);
  v2f a; a.x = p[0]; a.y = p[1];
  return a;
}

// B fragment: 4x16 tile of row-major W (leading dim ld), origin (k0,n0).
// lane: n = L%16 col, h = L/16 selects K pair; regs hold K=2h, 2h+1.
__device__ __forceinline__ v2f ldB(const float* W, int ld, int k0, int n0,
                                   int n, int h) {
  const float* p = W + (size_t)(k0 + 2 * h) * ld + (n0 + n);
  v2f b; b.x = p[0]; b.y = p[ld];
  return b;
}

// ---------------- tiny setup kernels ----------------
__global__ void k_flags_zero(int* flags) {
  int t = blockIdx.x * blockDim.x + threadIdx.x;
  if (t < U_N) flags[t] = 0;
}
__global__ void k_flags_set(const int* mask, int* flags) {
  int t = blockIdx.x * blockDim.x + threadIdx.x;
  if (t < MSK_N) flags[mask[t]] = 1;
}

// cumulative sums of relation weights along r
__global__ void k_cumsum(const float* g1u, const float* g1v, const float* g2u,
                         const float* g2v, float* c1u, float* c1v, float* c2u,
                         float* c2v) {
  int t = blockIdx.x * blockDim.x + threadIdx.x;
  if (t < 64 * 32) {
    float au = 0.f, av = 0.f;
    for (int r = 0; r < R_N; ++r) {
      au += g1u[r * 2048 + t]; av += g1v[r * 2048 + t];
      c1u[r * 2048 + t] = au;  c1v[r * 2048 + t] = av;
    }
  }
  if (t < 32 * 16) {
    float au = 0.f, av = 0.f;
    for (int r = 0; r < R_N; ++r) {
      au += g2u[r * 512 + t]; av += g2v[r * 512 + t];
      c2u[r * 512 + t] = au;  c2v[r * 512 + t] = av;
    }
  }
}

// row sums of S[r] -> safe inv sqrt.  One wave per row, float4 streaming.
__global__ void k_rowinv(const float* S, float* rowinv) {
  int wave = (blockIdx.x * blockDim.x + threadIdx.x) >> 5;
  int lane = threadIdx.x & 31;
  if (wave >= R_N * U_N) return;
  const float4* row = (const float4*)(S + (size_t)wave * I_N);
  float s = 0.f;
  for (int j = lane; j < I_N / 4; j += 32) {
    float4 v = row[j];
    s += v.x + v.y + v.z + v.w;
  }
#pragma unroll
  for (int off = 16; off; off >>= 1) s += __shfl_xor(s, off, 32);
  if (lane == 0) rowinv[wave] = (s > 0.f) ? (1.0f / sqrtf(s)) : 0.f;
}

// column partial sums over 256-row chunks (deterministic: fixed slots, no atomics)
__global__ void k_colsum_part(const float* S, float* colpart) {
  int t = blockIdx.x * blockDim.x + threadIdx.x;  // column id in [0, R*I)
  if (t >= R_N * I_N) return;
  int r = t >> 11, i = t & (I_N - 1);
  int chunk = blockIdx.y;                          // 32 chunks of 256 rows
  const float* base =
      S + (size_t)r * U_N * I_N + (size_t)chunk * 256 * I_N + i;
  float s = 0.f;
  for (int u = 0; u < 256; ++u) s += base[(size_t)u * I_N];
  colpart[(size_t)chunk * (R_N * I_N) + t] = s;
}
__global__ void k_colfin(const float* colpart, float* colinv) {
  int t = blockIdx.x * blockDim.x + threadIdx.x;
  if (t >= R_N * I_N) return;
  float s = 0.f;
  for (int c = 0; c < 32; ++c) s += colpart[(size_t)c * (R_N * I_N) + t];
  colinv[t] = (s > 0.f) ? (1.0f / sqrtf(s)) : 0.f;
}

// ---------------- fused 2-layer MLP: Out = leaky(X@W1 + b1) @ W2 + b2 ------
// N1 fixed at 32. One wave per 16-row strip. Optional row-zero flags on X.
__global__ __launch_bounds__(32) void k_mlp2(const float* X, const int* flags,
                                             const float* W1, int K1,
                                             const float* b1, const float* W2,
                                             int N2, const float* b2,
                                             float* Out) {
  __shared__ float t[16][33];
  int lane = threadIdx.x & 31;
  int n = lane & 15, h = lane >> 4;
  int row0 = blockIdx.x * 16;
  int m = n;  // A-row handled by this lane

  float fs = 1.f;
  if (flags) fs = flags[row0 + m] ? 0.f : 1.f;

  v8f c0 = {}; v8f c1 = {};
  for (int k = 0; k < K1; k += 4) {
    v2f a = ldA(X, K1, row0, k, m, h);
    a.x *= fs; a.y *= fs;
    c0 = wmma_f32(a, ldB(W1, 32, k, 0, n, h), c0);
    c1 = wmma_f32(a, ldB(W1, 32, k, 16, n, h), c1);
  }
  float bb0 = b1 ? b1[n] : 0.f;
  float bb1 = b1 ? b1[16 + n] : 0.f;
#pragma unroll
  for (int j = 0; j < 8; ++j) {
    int mm = j + 8 * h;
    float v0 = c0[j] + bb0; t[mm][n]      = v0 > 0.f ? v0 : 0.01f * v0;
    float v1 = c1[j] + bb1; t[mm][16 + n] = v1 > 0.f ? v1 : 0.01f * v1;
  }
  __syncthreads();

  for (int n0 = 0; n0 < N2; n0 += 16) {
    v8f c = {};
#pragma unroll
    for (int k = 0; k < 32; k += 4) {
      v2f a; a.x = t[m][k + 2 * h]; a.y = t[m][k + 2 * h + 1];
      c = wmma_f32(a, ldB(W2, N2, k, n0, n, h), c);
    }
    float bb = b2 ? b2[n0 + n] : 0.f;
#pragma unroll
    for (int j = 0; j < 8; ++j)
      Out[(size_t)(row0 + j + 8 * h) * N2 + n0 + n] = c[j] + bb;
  }
}

// ------- Out_r = scale_r[row] * (X @ W_r), r = blockIdx.y -----------------
__global__ __launch_bounds__(32) void k_gemm_scale(const float* X, int K,
                                                   const float* Wbase, int wStride,
                                                   const float* scaleBase, int sStride,
                                                   float* OutBase, int oStride,
                                                   int N) {
  int lane = threadIdx.x & 31;
  int n = lane & 15, h = lane >> 4;
  int tiles_n = N >> 4;
  int row0 = (blockIdx.x / tiles_n) * 16;
  int n0 = (blockIdx.x % tiles_n) * 16;
  int r = blockIdx.y;
  const float* W = Wbase + (size_t)r * wStride;
  const float* sc = scaleBase ? scaleBase + (size_t)r * sStride : nullptr;
  float* Out = OutBase + (size_t)r * oStride;

  v8f c = {};
  for (int k = 0; k < K; k += 4) {
    v2f a = ldA(X, K, row0, k, n, h);
    c = wmma_f32(a, ldB(W, N, k, n0, n, h), c);
  }
#pragma unroll
  for (int j = 0; j < 8; ++j) {
    int mm = row0 + j + 8 * h;
    float s = sc ? sc[mm] : 1.f;
    Out[(size_t)mm * N + n0 + n] = s * c[j];
  }
}

// ------- aggregation (split-K, 8 waves/block, dual accumulator chains) ----
// Out[j] = relu(b + sum_r asc_r[srow_j]*(A_r[srow_j,:]@H_r))
// A_r(m,k) = trans ? S[r][k][m] : S[r][m][k]. srow = idx ? idx[j] : j.
__global__ __launch_bounds__(AGG_NW * 32) void k_agg(
    const float* S, const float* Hbase, int hStride, const float* ascBase,
    int aStride, const float* bias, const int* idx, float* Out, int N, int Kd,
    int trans) {
  __shared__ float red[AGG_NW][16][32];
  int lane = threadIdx.x & 31;
  int wave = threadIdx.x >> 5;
  int n = lane & 15, h = lane >> 4;
  int row0 = blockIdx.x * 16;
  int m = n;
  int srow = idx ? idx[row0 + m] : (row0 + m);
  int tiles_n = N >> 4;
  int Kq = Kd / AGG_NW;
  int k0 = wave * Kq, k1 = k0 + Kq;

  v8f acc[2][2] = {};
  for (int r = 0; r < R_N; ++r) {
    const float* Sr = S + (size_t)r * U_N * I_N;
    const float* H = Hbase + (size_t)r * hStride;
    float asc = ascBase[(size_t)r * aStride + srow];
    if (trans) {
      for (int k = k0; k < k1; k += 8) {
        const float* p0 = Sr + (size_t)(k + 2 * h) * I_N + srow;
        const float* p1 = Sr + (size_t)(k + 4 + 2 * h) * I_N + srow;
        v2f a0, a1;
        a0.x = p0[0] * asc; a0.y = p0[I_N] * asc;
        a1.x = p1[0] * asc; a1.y = p1[I_N] * asc;
        for (int nt = 0; nt < tiles_n; ++nt) {
          acc[nt][0] = wmma_f32(a0, ldB(H, N, k, nt * 16, n, h), acc[nt][0]);
          acc[nt][1] = wmma_f32(a1, ldB(H, N, k + 4, nt * 16, n, h), acc[nt][1]);
        }
      }
    } else {
      for (int k = k0; k < k1; k += 8) {
        const float* p = Sr + (size_t)srow * I_N + (k + 2 * h);
        v2f a0, a1;
        a0.x = p[0] * asc; a0.y = p[1] * asc;
        a1.x = p[4] * asc; a1.y = p[5] * asc;
        for (int nt = 0; nt < tiles_n; ++nt) {
          acc[nt][0] = wmma_f32(a0, ldB(H, N, k, nt * 16, n, h), acc[nt][0]);
          acc[nt][1] = wmma_f32(a1, ldB(H, N, k + 4, nt * 16, n, h), acc[nt][1]);
        }
      }
    }
  }
  for (int nt = 0; nt < tiles_n; ++nt) {
#pragma unroll
    for (int j = 0; j < 8; ++j)
      red[wave][j + 8 * h][nt * 16 + n] = acc[nt][0][j] + acc[nt][1][j];
  }
  __syncthreads();
  int total = 16 * N;
  for (int e = threadIdx.x; e < total; e += AGG_NW * 32) {
    int mm = e / N, col = e - mm * N;
    float s = 0.f;
#pragma unroll
    for (int w = 0; w < AGG_NW; ++w) s += red[w][mm][col];
    s += bias[col];
    Out[(size_t)(row0 + mm) * N + col] = s > 0.f ? s : 0.f;
  }
}

// ------- score: relu(recon_u[mask] @ w1 + b1) @ w2 + b2  ([512,1]) --------
__global__ __launch_bounds__(32) void k_score(const float* recon_u,
                                              const int* mask, const float* w1,
                                              const float* b1, const float* w2,
                                              const float* b2, float* out) {
  __shared__ float t[16][65];
  int lane = threadIdx.x & 31;
  int n = lane & 15, h = lane >> 4;
  int row0 = blockIdx.x * 16;
  int m = n;
  int g = mask[row0 + m];
  const float* X = recon_u + (size_t)g * DIN_N;

  v8f c0 = {}, c1 = {}, c2 = {}, c3 = {};
  for (int k = 0; k < DIN_N; k += 4) {
    v2f a; a.x = X[k + 2 * h]; a.y = X[k + 2 * h + 1];
    c0 = wmma_f32(a, ldB(w1, 64, k, 0, n, h), c0);
    c1 = wmma_f32(a, ldB(w1, 64, k, 16, n, h), c1);
    c2 = wmma_f32(a, ldB(w1, 64, k, 32, n, h), c2);
    c3 = wmma_f32(a, ldB(w1, 64, k, 48, n, h), c3);
  }
#pragma unroll
  for (int j = 0; j < 8; ++j) {
    int mm = j + 8 * h;
    float v;
    v = c0[j] + b1[n];      t[mm][n]      = v > 0.f ? v : 0.f;
    v = c1[j] + b1[16 + n]; t[mm][16 + n] = v > 0.f ? v : 0.f;
    v = c2[j] + b1[32 + n]; t[mm][32 + n] = v > 0.f ? v : 0.f;
    v = c3[j] + b1[48 + n]; t[mm][48 + n] = v > 0.f ? v : 0.f;
  }
  __syncthreads();
  if (lane < 16) {
    float s = b2[0];
    for (int k = 0; k < 64; ++k) s += t[lane][k] * w2[k];
    out[row0 + lane] = s;
  }
}

// ---------------- host-side orchestration ----------------
extern "C" void kernel_launch(void* const* d_in, const int* in_sizes, int n_in,
                              void* d_out, int out_size, void* d_ws,
                              size_t ws_size, hipStream_t stream) {
  (void)in_sizes; (void)n_in; (void)out_size; (void)ws_size;
  const float* u_feat = (const float*)d_in[0];
  const float* v_feat = (const float*)d_in[1];
  const float* S      = (const float*)d_in[2];
  const int*   u_idx  = (const int*)d_in[3];
  const int*   v_idx  = (const int*)d_in[4];
  const int*   mask   = (const int*)d_in[5];
  const float* pw1    = (const float*)d_in[6];
  const float* pw2    = (const float*)d_in[7];
  const float* g1_uw  = (const float*)d_in[8];
  const float* g1_vw  = (const float*)d_in[9];
  const float* g1_b   = (const float*)d_in[10];
  const float* g2_uw  = (const float*)d_in[11];
  const float* g2_vw  = (const float*)d_in[12];
  const float* g2_b   = (const float*)d_in[13];
  const float* du_w   = (const float*)d_in[14];
  const float* dv_w   = (const float*)d_in[15];
  const float* dec_w1 = (const float*)d_in[16];
  const float* dec_b1 = (const float*)d_in[17];
  const float* dec_w2 = (const float*)d_in[18];
  const float* dec_b2 = (const float*)d_in[19];
  const float* inf_w1 = (const float*)d_in[20];
  const float* inf_b1 = (const float*)d_in[21];
  const float* inf_w2 = (const float*)d_in[22];
  const float* inf_b2 = (const float*)d_in[23];

  // workspace layout (floats); ~12.8 MB total
  float* ws      = (float*)d_ws;
  float* up      = ws;                          // [U,64]
  float* vp      = up + (size_t)U_N * 64;       // [I,64]
  float* rowinv  = vp + (size_t)I_N * 64;       // [R,U]
  float* colinv  = rowinv + (size_t)R_N * U_N;  // [R,I]
  float* c1u     = colinv + (size_t)R_N * I_N;  // [R,64,32]
  float* c1v     = c1u + (size_t)R_N * 64 * 32;
  float* c2u     = c1v + (size_t)R_N * 64 * 32; // [R,32,16]
  float* c2v     = c2u + (size_t)R_N * 32 * 16;
  int*   flags   = (int*)(c2v + (size_t)R_N * 32 * 16);  // [U]
  float* hv1     = (float*)(flags + U_N);       // [R,I,32]
  float* hu1     = hv1 + (size_t)R_N * I_N * 32;   // [R,U,32]
  float* zu1     = hu1 + (size_t)R_N * U_N * 32;   // [U,32]
  float* zv1     = zu1 + (size_t)U_N * 32;         // [I,32]
  float* zu2     = zv1 + (size_t)I_N * 32;         // [U,16]
  float* zv2     = zu2 + (size_t)U_N * 16;         // [I,16]
  float* colpart = zv2 + (size_t)I_N * 16;         // [32, R*I]
  float* hv2     = hv1;  // reuse: [R,I,16]
  float* hu2     = hu1;  // reuse: [R,U,16]
  float* uh      = zu1;  // reuse: [U,32]
  float* vh      = zv1;  // reuse: [I,32]

  float* recon_u = (float*)d_out;                 // [U,256]
  float* recon_v = recon_u + (size_t)U_N * DIN_N; // [I,256]
  float* score   = recon_v + (size_t)I_N * DIN_N; // [512]

  // 0) mask flags + cumulative relation weights + graph normalizers
  k_flags_zero<<<(U_N + 255) / 256, 256, 0, stream>>>(flags);
  k_flags_set<<<(MSK_N + 255) / 256, 256, 0, stream>>>(mask, flags);
  k_cumsum<<<8, 256, 0, stream>>>(g1_uw, g1_vw, g2_uw, g2_vw, c1u, c1v, c2u, c2v);
  k_rowinv<<<(R_N * U_N) / 8, 256, 0, stream>>>(S, rowinv);
  k_colsum_part<<<dim3((R_N * I_N) / 256, 32), 256, 0, stream>>>(S, colpart);
  k_colfin<<<(R_N * I_N) / 256, 256, 0, stream>>>(colpart, colinv);

  // 1) feature projections (mask-zeroed user rows)
  k_mlp2<<<U_N / 16, 32, 0, stream>>>(u_feat, flags, pw1, DIN_N, nullptr,
                                      pw2, 64, nullptr, up);
  k_mlp2<<<I_N / 16, 32, 0, stream>>>(v_feat, nullptr, pw1, DIN_N, nullptr,
                                      pw2, 64, nullptr, vp);

  // 2) encoder 1: pre-scaled per-relation transforms then aggregation
  k_gemm_scale<<<dim3((I_N / 16) * 2, R_N), 32, 0, stream>>>(
      vp, 64, c1v, 64 * 32, colinv, I_N, hv1, I_N * 32, 32);
  k_gemm_scale<<<dim3((U_N / 16) * 2, R_N), 32, 0, stream>>>(
      up, 64, c1u, 64 * 32, rowinv, U_N, hu1, U_N * 32, 32);
  k_agg<<<U_N / 16, AGG_NW * 32, 0, stream>>>(S, hv1, I_N * 32, rowinv, U_N,
                                              g1_b, nullptr, zu1, 32, I_N, 0);
  k_agg<<<I_N / 16, AGG_NW * 32, 0, stream>>>(S, hu1, U_N * 32, colinv, I_N,
                                              g1_b, nullptr, zv1, 32, U_N, 1);

  // 3) encoder 2 (gather rows via u/v index arrays)
  k_gemm_scale<<<dim3(I_N / 16, R_N), 32, 0, stream>>>(
      zv1, 32, c2v, 32 * 16, colinv, I_N, hv2, I_N * 16, 16);
  k_gemm_scale<<<dim3(U_N / 16, R_N), 32, 0, stream>>>(
      zu1, 32, c2u, 32 * 16, rowinv, U_N, hu2, U_N * 16, 16);
  k_agg<<<U_N / 16, AGG_NW * 32, 0, stream>>>(S, hv2, I_N * 16, rowinv, U_N,
                                              g2_b, u_idx, zu2, 16, I_N, 0);
  k_agg<<<I_N / 16, AGG_NW * 32, 0, stream>>>(S, hu2, U_N * 16, colinv, I_N,
                                              g2_b, v_idx, zv2, 16, U_N, 1);

  // 4) hidden projections
  k_gemm_scale<<<dim3((U_N / 16) * 2, 1), 32, 0, stream>>>(
      zu2, 16, du_w, 0, nullptr, 0, uh, 0, 32);
  k_gemm_scale<<<dim3((I_N / 16) * 2, 1), 32, 0, stream>>>(
      zv2, 16, dv_w, 0, nullptr, 0, vh, 0, 32);

  // 5) decoders -> recon_u / recon_v directly in d_out
  k_mlp2<<<U_N / 16, 32, 0, stream>>>(uh, nullptr, dec_w1, 32, dec_b1,
                                      dec_w2, DIN_N, dec_b2, recon_u);
  k_mlp2<<<I_N / 16, 32, 0, stream>>>(vh, nullptr, dec_w1, 32, dec_b1,
                                      dec_w2, DIN_N, dec_b2, recon_v);

  // 6) inference scores over masked rows
  k_score<<<MSK_N / 16, 32, 0, stream>>>(recon_u, mask, inf_w1, inf_b1,
                                         inf_w2, inf_b2, score);
}